// Mask_Loss_2705829397113
// MI455X (gfx1250) — compile-verified
//
#include <hip/hip_runtime.h>

typedef __attribute__((ext_vector_type(2))) float v2f;
typedef __attribute__((ext_vector_type(8))) float v8f;

// Problem shape, fixed by the reference's setup_inputs().
constexpr int B = 8, S = 512, D = 2640, M = 64;
constexpr float MARGIN = 15000.0f;

// ---------------------------------------------------------------------------
// Kernel 1: row norms. One wave32 per row.
//   rows [0, S)      -> ||input[B-1][s]||^2          -> ws[s]
//   rows [S, S+M)    -> ||target[B-1][idx[m]]||^2    -> ws[S+m]
// Also zeroes the scalar output (stream-ordered before kernel 2's atomics).
// Grid: 72 blocks x 256 threads = 576 waves exactly (S + M = 576).
// ---------------------------------------------------------------------------
__global__ void mask_loss_norms(const float* __restrict__ input,
                                const float* __restrict__ target,
                                const long long* __restrict__ mask,
                                float* __restrict__ ws,
                                float* __restrict__ out) {
    if (blockIdx.x == 0 && threadIdx.x == 0) out[0] = 0.0f;

    const int wave = (blockIdx.x * blockDim.x + threadIdx.x) >> 5;
    const int lane = threadIdx.x & 31;

    const float* row;
    if (wave < S) {
        row = input + ((size_t)(B - 1) * S + wave) * D;
    } else {
        const long long r = mask[(size_t)(B - 1) * M + (wave - S)];
        row = target + ((size_t)(B - 1) * S + (size_t)r) * D;
    }

    float acc = 0.0f;
    for (int i = lane; i < D; i += 32) {
        const float v = row[i];
        acc = fmaf(v, v, acc);
    }
    #pragma unroll
    for (int off = 16; off; off >>= 1) acc += __shfl_down(acc, off, 32);
    if (lane == 0) ws[wave] = acc;
}

// ---------------------------------------------------------------------------
// Kernel 2: one wave32 per 16x16 tile of G[m][s] = reh[m] . inp[s], computed
// with V_WMMA_F32_16X16X4_F32 over the K = D = 2640 dimension.
//
// A (16x4 fp32) VGPR layout: lane = 16*half + r holds A[r][2*half + {0,1}]
// B ( 4x16 fp32)            : lane = 16*half + n holds B[2*half + {0,1}][n]
// C/D (16x16 f32, 8 VGPRs)  : lane = 16*half + n, VGPR j holds C[j + 8*half][n]
//
// Epilogue: sq = ||a||^2 + ||b||^2 - 2G, hinge select, wave reduce, 1 atomic.
// Grid: 16 blocks x 256 threads = 128 waves = 128 tiles (4 Mtiles x 32 Stiles).
// ---------------------------------------------------------------------------
__global__ void mask_loss_tiles(const float* __restrict__ input,
                                const float* __restrict__ target,
                                const long long* __restrict__ mask,
                                const float* __restrict__ ws,
                                float* __restrict__ out) {
    const int lane   = threadIdx.x & 31;
    const int wave   = threadIdx.x >> 5;
    const int tile   = blockIdx.x * (blockDim.x >> 5) + wave;  // 0..127
    const int tileM  = tile >> 5;   // 0..3   (M/16)
    const int tileS  = tile & 31;   // 0..31  (S/16)
    const int half   = lane >> 4;   // 0 or 1
    const int r      = lane & 15;   // row (A) / col (B) within tile

    // A rows come from the masked gather of target[B-1]; B rows from input[B-1].
    const long long aRow = mask[(size_t)(B - 1) * M + tileM * 16 + r];
    const float* aPtr = target + ((size_t)(B - 1) * S + (size_t)aRow) * D + 2 * half;
    const float* bPtr = input  + ((size_t)(B - 1) * S + (size_t)(tileS * 16 + r)) * D + 2 * half;

    // Two accumulators break the WMMA->WMMA RAW dependency chain.
    v8f c0 = {};
    v8f c1 = {};
    #pragma unroll 2
    for (int k = 0; k < D; k += 8) {
        const v2f a0 = *(const v2f*)(aPtr + k);
        const v2f b0 = *(const v2f*)(bPtr + k);
        const v2f a1 = *(const v2f*)(aPtr + k + 4);
        const v2f b1 = *(const v2f*)(bPtr + k + 4);
        c0 = __builtin_amdgcn_wmma_f32_16x16x4_f32(false, a0, false, b0,
                                                   (short)0, c0, false, false);
        c1 = __builtin_amdgcn_wmma_f32_16x16x4_f32(false, a1, false, b1,
                                                   (short)0, c1, false, false);
    }
    const v8f c = c0 + c1;

    // Epilogue per lane: this lane owns column n = r of the tile, rows j+8*half.
    const int sGlob = tileS * 16 + r;
    const float nB = ws[sGlob];

    float acc = 0.0f;
    #pragma unroll
    for (int j = 0; j < 8; ++j) {
        const int mGlob = tileM * 16 + j + 8 * half;
        const float nA = ws[S + mGlob];
        const long long im = mask[(size_t)(B - 1) * M + mGlob];
        const float sq = nA + nB - 2.0f * c[j];
        const long long d = im - (long long)sGlob;
        const bool lab = (d <= 1) && (d >= -1);
        acc += lab ? sq : fmaxf(0.0f, MARGIN - sq);
    }

    #pragma unroll
    for (int off = 16; off; off >>= 1) acc += __shfl_down(acc, off, 32);

    if (lane == 0) {
        constexpr double scaled = 1.0 / ((double)S * (double)(S - 1)
                                         * (double)M * (double)(B * 100));
        atomicAdd(out, acc * (float)scaled);
    }
}

extern "C" void kernel_launch(void* const* d_in, const int* in_sizes, int n_in,
                              void* d_out, int out_size, void* d_ws, size_t ws_size,
                              hipStream_t stream) {
    const float*     input  = (const float*)d_in[0];
    const long long* mask   = (const long long*)d_in[1];   // int64 indices
    const float*     target = (const float*)d_in[2];
    float*           out    = (float*)d_out;
    float*           ws     = (float*)d_ws;                // 576 floats used

    // Kernel 1: 576 waves (one per row norm) + zero the output scalar.
    mask_loss_norms<<<dim3(72), dim3(256), 0, stream>>>(input, target, mask, ws, out);
    // Kernel 2: 128 waves, one 16x16 WMMA tile each.
    mask_loss_tiles<<<dim3(16), dim3(256), 0, stream>>>(input, target, mask, ws, out);
}